// LoRAMoE_49520972923178
// MI455X (gfx1250) — compile-verified
//
#include <hip/hip_runtime.h>

// LoRA-MoE fused kernel for gfx1250 (MI455X), fp32 via V_WMMA_F32_16X16X4_F32,
// with async global->LDS staging (ASYNCcnt pipeline) of the activation tile.
//
// x:      [4, 4096, 1024] f32
// W_down: [4, 1024, 64]   f32  (per-expert, row-major [d, a])
// W_up:   [4, 64, 1024]   f32  (flattened == [256, 1024] K-major for fused up-GEMM)
// W_gate: [1024, 4] f32,  b_gate: [4] f32
// out:    [4, 4096, 1024] f32
//
// Block = 256 threads (8 wave32) handles MT=32 tokens; X streamed in 4 K-chunks
// of 256 (double-buffered LDS), down-accumulators held in registers across chunks.

typedef float v2f __attribute__((ext_vector_type(2)));
typedef float v8f __attribute__((ext_vector_type(8)));

#define D_MODEL 1024
#define ADIM    64
#define NE      4
#define MT      32       // tokens per block (2 row tiles -> B fragment reused 2x)
#define CK      256      // k-chunk length
#define NCHUNK  (D_MODEL / CK)
#define CS      260      // chunk row stride: 260 % 64 == 4 -> conflict-free A reads, even -> 8B aligned
#define HS      260      // H' row stride, same property

__global__ __launch_bounds__(256) void lora_moe_kernel(
    const float* __restrict__ x,
    const float* __restrict__ Wd,
    const float* __restrict__ Wu,
    const float* __restrict__ Wg,
    const float* __restrict__ bg,
    float* __restrict__ out)
{
    __shared__ float Xc[2][MT * CS];     // double-buffered X chunk [32 x 256]
    __shared__ float Hs[MT * HS];        // gated down-projection H' [32 x 256]
    __shared__ float gateS[MT * NE];     // softmax gate weights

    float* part = Hs;                    // gate partial scratch aliases Hs (used before Hs is written)

    const int tid  = threadIdx.x;
    const int wave = tid >> 5;
    const int lane = tid & 31;
    const int col  = lane & 15;          // N index (and A-fragment row within a row tile)
    const int hi   = lane >> 4;          // lane half
    const int koff = hi * 2;             // K sub-offset per f32 WMMA fragment layout
    const long tok0 = (long)blockIdx.x * MT;
    const float* xsrc = x + tok0 * D_MODEL;

    // ---- async stage one K-chunk of X into LDS buffer `buf` (16 b64 ops / thread / chunk) ----
    auto issue_chunk = [&](int c, int buf) {
        unsigned lds0 = (unsigned)(size_t)&Xc[buf][0];
        for (int i = tid; i < MT * (CK / 2); i += 256) {   // 128 b64 chunks per row
            int row = i >> 7;
            int c2  = i & 127;
            unsigned loff = lds0 + (unsigned)((row * CS + c2 * 2) * 4);
            unsigned long long ga =
                (unsigned long long)(size_t)(xsrc + (size_t)row * D_MODEL + c * CK + c2 * 2);
            asm volatile("global_load_async_to_lds_b64 %0, %1, off"
                         :: "v"(loff), "v"(ga) : "memory");
        }
    };

    // warm L2 with weights (global_prefetch_b8)
    __builtin_prefetch(Wu + (size_t)tid * 1024, 0, 1);
    __builtin_prefetch(Wd + (size_t)tid * 1024, 0, 1);

    issue_chunk(0, 0);
    issue_chunk(1, 1);

    // per-thread gate partial:  p = tid>>1 in [0,128): p = t*4+e ; sub = tid&1 splits K 2-way
    const int gp   = tid >> 1;
    const int gsub = tid & 1;
    const int gt   = gp >> 2;
    const int ge   = gp & 3;
    float gacc = 0.f;

    // down-phase accumulators: wave owns col tiles ct0=2w, ct1=2w+1 for BOTH row tiles
    const int ct0 = wave * 2, ct1 = ct0 + 1;
    const int e0 = ct0 >> 2, n00 = (ct0 & 3) * 16;
    const int e1 = ct1 >> 2, n01 = (ct1 & 3) * 16;
    const float* B0d = Wd + (long)e0 * D_MODEL * ADIM + n00 + col;
    const float* B1d = Wd + (long)e1 * D_MODEL * ADIM + n01 + col;
    v8f acc00 = {}, acc01 = {}, acc10 = {}, acc11 = {};

    for (int c = 0; c < NCHUNK; ++c) {
        const int buf = c & 1;
        if (c < NCHUNK - 1)
            asm volatile("s_wait_asynccnt 0x10" ::: "memory");   // own older chunk done
        else
            asm volatile("s_wait_asynccnt 0x0" ::: "memory");
        __syncthreads();

        // gate partial over this chunk (reads broadcast-friendly LDS rows)
        {
            const float* xr = &Xc[buf][gt * CS + gsub * 128];
            const float* wg = Wg + (c * CK + gsub * 128) * NE + ge;
            #pragma unroll 8
            for (int k = 0; k < 128; ++k)
                gacc = fmaf(xr[k], wg[k * NE], gacc);
        }

        // down wmma over this chunk
        {
            const float* a0r = &Xc[buf][col * CS];
            const float* a1r = &Xc[buf][(16 + col) * CS];
            #pragma unroll 2
            for (int kl = 0; kl < CK; kl += 4) {
                int kb  = kl + koff;
                int kgb = c * CK + kb;
                v2f a0 = *(const v2f*)&a0r[kb];
                v2f a1 = *(const v2f*)&a1r[kb];
                v2f b0; b0.x = B0d[kgb * ADIM]; b0.y = B0d[(kgb + 1) * ADIM];
                v2f b1; b1.x = B1d[kgb * ADIM]; b1.y = B1d[(kgb + 1) * ADIM];
                acc00 = __builtin_amdgcn_wmma_f32_16x16x4_f32(false, a0, false, b0, (short)0, acc00, false, false);
                acc10 = __builtin_amdgcn_wmma_f32_16x16x4_f32(false, a1, false, b0, (short)0, acc10, false, false);
                acc01 = __builtin_amdgcn_wmma_f32_16x16x4_f32(false, a0, false, b1, (short)0, acc01, false, false);
                acc11 = __builtin_amdgcn_wmma_f32_16x16x4_f32(false, a1, false, b1, (short)0, acc11, false, false);
            }
        }
        __syncthreads();                       // everyone done reading buf before overwrite
        if (c + 2 < NCHUNK) issue_chunk(c + 2, buf);
    }

    // ---- gate reduce + softmax ----
    part[tid] = gacc;                          // part[p*2 + sub]
    __syncthreads();
    if (tid < MT * NE) {                       // 128 logits
        int e = tid & 3;
        gateS[tid] = part[tid * 2] + part[tid * 2 + 1] + bg[e];
    }
    __syncthreads();
    if (tid < MT) {
        float l0 = gateS[tid * 4 + 0], l1 = gateS[tid * 4 + 1];
        float l2 = gateS[tid * 4 + 2], l3 = gateS[tid * 4 + 3];
        float m  = fmaxf(fmaxf(l0, l1), fmaxf(l2, l3));
        float x0 = __expf(l0 - m), x1 = __expf(l1 - m);
        float x2 = __expf(l2 - m), x3 = __expf(l3 - m);
        float inv = 1.f / (x0 + x1 + x2 + x3);
        gateS[tid * 4 + 0] = x0 * inv;
        gateS[tid * 4 + 1] = x1 * inv;
        gateS[tid * 4 + 2] = x2 * inv;
        gateS[tid * 4 + 3] = x3 * inv;
    }
    __syncthreads();

    // ---- fold gate into H' and stage to LDS ----
    #pragma unroll
    for (int r = 0; r < 8; ++r) {
        int M0 = r + hi * 8;
        int M1 = 16 + M0;
        Hs[M0 * HS + e0 * ADIM + n00 + col] = acc00[r] * gateS[M0 * NE + e0];
        Hs[M1 * HS + e0 * ADIM + n00 + col] = acc10[r] * gateS[M1 * NE + e0];
        Hs[M0 * HS + e1 * ADIM + n01 + col] = acc01[r] * gateS[M0 * NE + e1];
        Hs[M1 * HS + e1 * ADIM + n01 + col] = acc11[r] * gateS[M1 * NE + e1];
    }
    __syncthreads();

    // ---- up phase: out[32,1024] = H'[32,256] @ Wu_flat[256,1024] ----
    #pragma unroll 1
    for (int p = 0; p < 4; ++p) {
        const int n0 = wave * 128 + p * 32;    // ct0 = wave*8 + 2p -> n0 = ct0*16
        const float* B0 = Wu + n0 + col;
        const float* B1 = Wu + n0 + 16 + col;
        const float* a0r = &Hs[col * HS];
        const float* a1r = &Hs[(16 + col) * HS];

        v8f u00 = {}, u01 = {}, u10 = {}, u11 = {};
        #pragma unroll 2
        for (int k = 0; k < NE * ADIM; k += 4) {
            int kb = k + koff;
            v2f a0 = *(const v2f*)&a0r[kb];
            v2f a1 = *(const v2f*)&a1r[kb];
            v2f b0; b0.x = B0[kb * D_MODEL]; b0.y = B0[(kb + 1) * D_MODEL];
            v2f b1; b1.x = B1[kb * D_MODEL]; b1.y = B1[(kb + 1) * D_MODEL];
            u00 = __builtin_amdgcn_wmma_f32_16x16x4_f32(false, a0, false, b0, (short)0, u00, false, false);
            u10 = __builtin_amdgcn_wmma_f32_16x16x4_f32(false, a1, false, b0, (short)0, u10, false, false);
            u01 = __builtin_amdgcn_wmma_f32_16x16x4_f32(false, a0, false, b1, (short)0, u01, false, false);
            u11 = __builtin_amdgcn_wmma_f32_16x16x4_f32(false, a1, false, b1, (short)0, u11, false, false);
        }
        #pragma unroll
        for (int r = 0; r < 8; ++r) {
            int M0 = r + hi * 8;
            int M1 = 16 + M0;
            out[(tok0 + M0) * D_MODEL + n0 + col]      = u00[r];
            out[(tok0 + M1) * D_MODEL + n0 + col]      = u10[r];
            out[(tok0 + M0) * D_MODEL + n0 + 16 + col] = u01[r];
            out[(tok0 + M1) * D_MODEL + n0 + 16 + col] = u11[r];
        }
    }
}

extern "C" void kernel_launch(void* const* d_in, const int* in_sizes, int n_in,
                              void* d_out, int out_size, void* d_ws, size_t ws_size,
                              hipStream_t stream) {
    const float* x  = (const float*)d_in[0];
    const float* Wd = (const float*)d_in[1];
    const float* Wu = (const float*)d_in[2];
    const float* Wg = (const float*)d_in[3];
    const float* bg = (const float*)d_in[4];
    float* out = (float*)d_out;

    const int tokens = in_sizes[0] / D_MODEL;   // 4*4096 = 16384
    dim3 grid(tokens / MT);                     // 512 blocks
    lora_moe_kernel<<<grid, 256, 0, stream>>>(x, Wd, Wu, Wg, bg, out);
}